// BridgeoutFcLayer_42966852829428
// MI455X (gfx1250) — compile-verified
//
#include <hip/hip_runtime.h>
#include <hip/hip_bf16.h>
#include <math.h>

typedef __attribute__((ext_vector_type(2))) float v2f;
typedef __attribute__((ext_vector_type(8))) float v8f;

#define BS    64
#define IN_F  1024
#define OUT_F 1024

// ---------------------------------------------------------------------------
// Kernel A: dense fp32 GEMM  out[b,o] = sum_i x[b,i]*w[i,o] + bias[o]
// via V_WMMA_F32_16X16X4_F32. One wave (32 threads) owns one 16x16 C tile.
// K loop: 1024 / 4 = 256 WMMA issues per wave. EXEC is all-1s (full wave,
// no divergence), as WMMA requires.
//
// A operand (16x4 f32, 2 VGPRs): lanes 0-15: M=lane, v0=K0, v1=K1;
//                                lanes 16-31: M=lane-16, v0=K2, v1=K3.
// B operand (4x16 f32, 2 VGPRs): lanes 0-15: N=lane, v0=K0 row, v1=K1 row;
//                                lanes 16-31: N=lane-16, v0=K2, v1=K3.
// C/D (16x16 f32, 8 VGPRs): VGPR j: lanes 0-15 -> M=j, lanes 16-31 -> M=j+8.
// ---------------------------------------------------------------------------
__global__ __launch_bounds__(32)
void dense_wmma_kernel(const float* __restrict__ x,
                       const float* __restrict__ w,
                       const float* __restrict__ bias,
                       float* __restrict__ out) {
    const int lane   = threadIdx.x;      // 0..31
    const int half   = lane >> 4;        // 0 or 1
    const int mn     = lane & 15;
    const int m_base = blockIdx.y * 16;  // batch tile
    const int n_base = blockIdx.x * 16;  // output-feature tile

    // Per-lane base pointers for the A / B fragments.
    const float* arow = x + (size_t)(m_base + mn) * IN_F + 2 * half;
    const float* bcol = w + (size_t)(2 * half) * OUT_F + (n_base + mn);

    v8f c = {};
    for (int k = 0; k < IN_F; k += 4) {
        v2f a, b;
        a.x = arow[k];                              // K = k + 2*half
        a.y = arow[k + 1];                          // K = k + 2*half + 1
        b.x = bcol[(size_t)k * OUT_F];              // row K = k + 2*half
        b.y = bcol[(size_t)k * OUT_F + OUT_F];      // row K = k + 2*half + 1
        c = __builtin_amdgcn_wmma_f32_16x16x4_f32(
                /*neg_a=*/false, a, /*neg_b=*/false, b,
                /*c_mod=*/(short)0, c, /*reuse_a=*/false, /*reuse_b=*/false);
    }

    const float bv = bias[n_base + mn];
#pragma unroll
    for (int j = 0; j < 8; ++j) {
        const int row = m_base + j + half * 8;
        out[(size_t)row * OUT_F + (n_base + mn)] = c[j] + bv;
    }
}

// ---------------------------------------------------------------------------
// Kernel B: HBM-bound streaming term
//   out[b,o] += sum_i x[b,i] * (|w[i,o]| + 1e-15) * noise[b,i,o]
// One block = one batch b x one 256-wide o tile. Each thread owns one o.
// noise is read once -> non-temporal loads (don't pollute L2; w wants to
// stay L2-resident since it's reused by all 64 batches). x row staged in LDS.
// ---------------------------------------------------------------------------
__global__ __launch_bounds__(256)
void noise_stream_kernel(const float* __restrict__ x,
                         const float* __restrict__ w,
                         const float* __restrict__ noise,
                         float* __restrict__ out) {
    __shared__ float xs[IN_F];
    const int b = blockIdx.y;
    const int o = blockIdx.x * 256 + threadIdx.x;

    for (int i = threadIdx.x; i < IN_F; i += 256)
        xs[i] = x[(size_t)b * IN_F + i];
    __syncthreads();

    const float* np = noise + (size_t)b * IN_F * OUT_F + o;
    const float* wp = w + o;

    float acc0 = 0.f, acc1 = 0.f, acc2 = 0.f, acc3 = 0.f;
    float acc4 = 0.f, acc5 = 0.f, acc6 = 0.f, acc7 = 0.f;

#pragma unroll 1
    for (int i = 0; i < IN_F; i += 8) {
        // 8 independent streams of loads -> deep MLP to hide HBM latency.
        float n0 = __builtin_nontemporal_load(np + (size_t)(i + 0) * OUT_F);
        float n1 = __builtin_nontemporal_load(np + (size_t)(i + 1) * OUT_F);
        float n2 = __builtin_nontemporal_load(np + (size_t)(i + 2) * OUT_F);
        float n3 = __builtin_nontemporal_load(np + (size_t)(i + 3) * OUT_F);
        float n4 = __builtin_nontemporal_load(np + (size_t)(i + 4) * OUT_F);
        float n5 = __builtin_nontemporal_load(np + (size_t)(i + 5) * OUT_F);
        float n6 = __builtin_nontemporal_load(np + (size_t)(i + 6) * OUT_F);
        float n7 = __builtin_nontemporal_load(np + (size_t)(i + 7) * OUT_F);

        float w0 = wp[(size_t)(i + 0) * OUT_F];
        float w1 = wp[(size_t)(i + 1) * OUT_F];
        float w2 = wp[(size_t)(i + 2) * OUT_F];
        float w3 = wp[(size_t)(i + 3) * OUT_F];
        float w4 = wp[(size_t)(i + 4) * OUT_F];
        float w5 = wp[(size_t)(i + 5) * OUT_F];
        float w6 = wp[(size_t)(i + 6) * OUT_F];
        float w7 = wp[(size_t)(i + 7) * OUT_F];

        acc0 = fmaf(xs[i + 0] * (fabsf(w0) + 1e-15f), n0, acc0);
        acc1 = fmaf(xs[i + 1] * (fabsf(w1) + 1e-15f), n1, acc1);
        acc2 = fmaf(xs[i + 2] * (fabsf(w2) + 1e-15f), n2, acc2);
        acc3 = fmaf(xs[i + 3] * (fabsf(w3) + 1e-15f), n3, acc3);
        acc4 = fmaf(xs[i + 4] * (fabsf(w4) + 1e-15f), n4, acc4);
        acc5 = fmaf(xs[i + 5] * (fabsf(w5) + 1e-15f), n5, acc5);
        acc6 = fmaf(xs[i + 6] * (fabsf(w6) + 1e-15f), n6, acc6);
        acc7 = fmaf(xs[i + 7] * (fabsf(w7) + 1e-15f), n7, acc7);
    }

    const float acc = ((acc0 + acc1) + (acc2 + acc3)) +
                      ((acc4 + acc5) + (acc6 + acc7));
    out[(size_t)b * OUT_F + o] += acc;   // unique owner of (b,o): no races
}

extern "C" void kernel_launch(void* const* d_in, const int* in_sizes, int n_in,
                              void* d_out, int out_size, void* d_ws, size_t ws_size,
                              hipStream_t stream) {
    const float* x     = (const float*)d_in[0];  // (64, 1024)
    const float* w     = (const float*)d_in[1];  // (1024, 1024)
    const float* bias  = (const float*)d_in[2];  // (1024,)
    const float* noise = (const float*)d_in[3];  // (64, 1024, 1024)
    float* out = (float*)d_out;                  // (64, 1024)

    // Kernel A fully overwrites d_out (dense GEMM + bias), then kernel B
    // accumulates the streamed noise term. Same stream -> ordered, and
    // every replay recomputes d_out from scratch (deterministic).
    dense_wmma_kernel<<<dim3(OUT_F / 16, BS / 16), 32, 0, stream>>>(x, w, bias, out);
    noise_stream_kernel<<<dim3(OUT_F / 256, BS), 256, 0, stream>>>(x, w, noise, out);
}